// AttentionPooling_20796231647300
// MI455X (gfx1250) — compile-verified
//
#include <hip/hip_runtime.h>
#include <hip/hip_bf16.h>
#include <math.h>
#include <stdint.h>

// ---------------------------------------------------------------------------
// AttentionPooling for MI455X (gfx1250), wave32 + V_WMMA_F32_16X16X4_F32.
// Flash-style fusion: x read from HBM exactly once per element.
// Staging uses gfx1250 async global->LDS (ASYNCcnt) instead of VGPR bounce.
// ---------------------------------------------------------------------------

typedef __attribute__((ext_vector_type(2))) float v2f;
typedef __attribute__((ext_vector_type(8))) float v8f;

#define BB    16                  // batch
#define NN    8192                // sequence length
#define DIN   512                 // d_in
#define DQ    16                  // d_out (query count) == WMMA M
#define NC    128                 // sequence rows per chunk / block
#define CPB   (NN / NC)           // 64 chunks per batch element
#define XP    516                 // LDS row pitch for x and W (bank-conflict pad)
#define SP    132                 // LDS row pitch for score rows
#define SCALE 0.044194173824159216f  // 1/sqrt(512)

// LDS partition (float offsets)
#define OFF_XS 0
#define OFF_WS (OFF_XS + NC * XP)       // x chunk: 128 x 516
#define OFF_SS (OFF_WS + DQ * XP)       // W:       16 x 516
#define OFF_PM (OFF_SS + DQ * SP)       // scores:  16 x 132
#define OFF_PS (OFF_PM + DQ * 16)       // partial max 16x16
#define OFF_M  (OFF_PS + DQ * 16)       // partial sum 16x16
#define OFF_L  (OFF_M + DQ)             // row max   [16]
#define LDS_FLOATS (OFF_L + DQ)         // row sum   [16]  -> total 76960 f = 307,840 B (< 320 KB)

// One 16-byte async global->LDS transfer per lane (gfx1250 ASYNCcnt path).
__device__ __forceinline__ void async_g2l_b128(const void* gaddr, const void* ldsptr) {
    const unsigned long long g = (unsigned long long)(uintptr_t)gaddr;
    const unsigned l = (unsigned)(uintptr_t)ldsptr;   // low 32 bits = LDS offset
    asm volatile("global_load_async_to_lds_b128 %0, %1, off"
                 :: "v"(l), "v"(g) : "memory");
}

__device__ __forceinline__ void wait_async_all() {
    asm volatile("s_wait_asynccnt 0x0" ::: "memory");
}

__global__ __launch_bounds__(256) void attnpool_chunk_kernel(
    const float* __restrict__ x, const float* __restrict__ w,
    float* __restrict__ Opart, float* __restrict__ Mpart, float* __restrict__ Lpart)
{
    extern __shared__ float lds[];
    float* Xs = lds + OFF_XS;
    float* Ws = lds + OFF_WS;
    float* Ss = lds + OFF_SS;
    float* Pm = lds + OFF_PM;
    float* Ps = lds + OFF_PS;
    float* Mr = lds + OFF_M;
    float* Lr = lds + OFF_L;

    const int blk  = blockIdx.x;
    const int b    = blk / CPB;
    const int c    = blk % CPB;
    const int tid  = threadIdx.x;
    const int wave = tid >> 5;     // 0..7
    const int lane = tid & 31;
    const int half = lane >> 4;    // 0/1 : which half-wave
    const int l15  = lane & 15;

    // ---- stage W (16 x 512) into LDS: async global->LDS, 16B per lane ----
    for (int i = tid; i < DQ * (DIN / 4); i += 256) {
        const int row = i >> 7, c4 = i & 127;
        async_g2l_b128(w + (size_t)row * DIN + c4 * 4, &Ws[row * XP + c4 * 4]);
    }
    // ---- stage x chunk (128 x 512 = 256 KB) into LDS: async global->LDS ----
    const float* xb = x + ((size_t)b * NN + (size_t)c * NC) * DIN;
    for (int i = tid; i < NC * (DIN / 4); i += 256) {
        const int row = i >> 7, c4 = i & 127;
        async_g2l_b128(xb + (size_t)row * DIN + c4 * 4, &Xs[row * XP + c4 * 4]);
    }
    wait_async_all();      // each wave drains its own ASYNCcnt
    __syncthreads();       // then cross-wave visibility

    // ---- Phase A: S(16q x 16n) = W(16x512) * Xchunk^T, one tile per wave ----
    // A frag: a[g] = W[l15][k + g + 2*half]      (contiguous float2 per lane)
    // B frag: b[g] = x[n0+l15][k + g + 2*half]   (contiguous float2 per lane)
    const int n0 = wave * 16;
    const float* ap = Ws + l15 * XP + 2 * half;
    const float* bp = Xs + (n0 + l15) * XP + 2 * half;
    v8f sacc = {0.f, 0.f, 0.f, 0.f, 0.f, 0.f, 0.f, 0.f};
#pragma unroll 4
    for (int k = 0; k < DIN; k += 4) {
        const v2f a  = *reinterpret_cast<const v2f*>(ap + k);
        const v2f bb = *reinterpret_cast<const v2f*>(bp + k);
        sacc = __builtin_amdgcn_wmma_f32_16x16x4_f32(
            false, a, false, bb, (short)0, sacc, false, false);
    }
    // D frag: sacc[g] = S[g + 8*half][l15]
#pragma unroll
    for (int g = 0; g < 8; ++g)
        Ss[(g + 8 * half) * SP + n0 + l15] = sacc[g] * SCALE;
    __syncthreads();

    // ---- Phase B: per-q chunk max, exp in place, per-q chunk sum ----
    {
        const int q = tid >> 4, j = tid & 15;
        float pm = -3.4e38f;
#pragma unroll
        for (int e = 0; e < 8; ++e) pm = fmaxf(pm, Ss[q * SP + j * 8 + e]);
        Pm[q * 16 + j] = pm;
    }
    __syncthreads();
    if (tid < DQ) {
        float m = -3.4e38f;
        for (int j = 0; j < 16; ++j) m = fmaxf(m, Pm[tid * 16 + j]);
        Mr[tid] = m;
    }
    __syncthreads();
    {
        const int q = tid >> 4, j = tid & 15;
        const float m = Mr[q];
        float ps = 0.f;
#pragma unroll
        for (int e = 0; e < 8; ++e) {
            const int idx = q * SP + j * 8 + e;
            const float p = __expf(Ss[idx] - m);
            Ss[idx] = p;
            ps += p;
        }
        Ps[q * 16 + j] = ps;
    }
    __syncthreads();
    if (tid < DQ) {
        float s = 0.f;
        for (int j = 0; j < 16; ++j) s += Ps[tid * 16 + j];
        Lr[tid] = s;
    }
    __syncthreads();

    // ---- Phase C: O(16q x 64d slice per wave) = P(16x128) * Xchunk(128x512) ----
    // A frag: a[g] = P[l15][n + g + 2*half]                (contiguous float2)
    // B frag: b[g] = x[n + g + 2*half][d0 + dt*16 + l15]   (two scalar LDS loads)
    v8f o[4];
#pragma unroll
    for (int dt = 0; dt < 4; ++dt)
        o[dt] = (v8f){0.f, 0.f, 0.f, 0.f, 0.f, 0.f, 0.f, 0.f};
    const float* sp2 = Ss + l15 * SP + 2 * half;
    const int d0 = wave * 64;
#pragma unroll 2
    for (int n = 0; n < NC; n += 4) {
        const v2f a = *reinterpret_cast<const v2f*>(sp2 + n);
        const float* xr0 = Xs + (n + 2 * half) * XP + d0 + l15;
        const float* xr1 = xr0 + XP;
#pragma unroll
        for (int dt = 0; dt < 4; ++dt) {
            v2f bb;
            bb.x = xr0[dt * 16];
            bb.y = xr1[dt * 16];
            o[dt] = __builtin_amdgcn_wmma_f32_16x16x4_f32(
                false, a, false, bb, (short)0, o[dt], false, false);
        }
    }

    // ---- write per-chunk partials: O[blk][16][512], m[blk][16], l[blk][16] ----
    float* Ob = Opart + (size_t)blk * DQ * DIN;
#pragma unroll
    for (int dt = 0; dt < 4; ++dt)
#pragma unroll
        for (int g = 0; g < 8; ++g)
            Ob[(g + 8 * half) * DIN + d0 + dt * 16 + l15] = o[dt][g];
    if (tid < DQ) {
        Mpart[blk * DQ + tid] = Mr[tid];
        Lpart[blk * DQ + tid] = Lr[tid];
    }
}

// ---------------------------------------------------------------------------
// Cross-chunk log-sum-exp combine: one block per (b, q).
// out[b][q][d] = sum_c e^{m_c - m*} O_c[q][d] / sum_c e^{m_c - m*} l_c
// ---------------------------------------------------------------------------
__global__ __launch_bounds__(128) void attnpool_combine_kernel(
    const float* __restrict__ Opart, const float* __restrict__ Mpart,
    const float* __restrict__ Lpart, float* __restrict__ out)
{
    __shared__ float wgt[CPB];
    __shared__ float inv;
    const int b   = blockIdx.x >> 4;
    const int q   = blockIdx.x & 15;
    const int tid = threadIdx.x;

    if (tid == 0) {
        float m = -3.4e38f;
        for (int cc = 0; cc < CPB; ++cc)
            m = fmaxf(m, Mpart[(b * CPB + cc) * DQ + q]);
        float l = 0.f;
        for (int cc = 0; cc < CPB; ++cc) {
            const float wv = __expf(Mpart[(b * CPB + cc) * DQ + q] - m);
            wgt[cc] = wv;
            l += wv * Lpart[(b * CPB + cc) * DQ + q];
        }
        inv = 1.0f / l;
    }
    __syncthreads();

    for (int d = tid; d < DIN; d += 128) {
        float acc = 0.f;
        for (int cc = 0; cc < CPB; ++cc)
            acc += wgt[cc] * Opart[((size_t)(b * CPB + cc) * DQ + q) * DIN + d];
        out[((size_t)b * DQ + q) * DIN + d] = acc * inv;
    }
}

// ---------------------------------------------------------------------------
extern "C" void kernel_launch(void* const* d_in, const int* in_sizes, int n_in,
                              void* d_out, int out_size, void* d_ws, size_t ws_size,
                              hipStream_t stream) {
    const float* x = (const float*)d_in[0];   // [16, 8192, 512] f32
    const float* w = (const float*)d_in[1];   // [1, 16, 512] f32
    float* out = (float*)d_out;               // [16, 16, 512] f32

    // workspace: O partials [1024][16][512], then m [1024][16], l [1024][16]
    float* Opart = (float*)d_ws;
    float* Mpart = Opart + (size_t)BB * CPB * DQ * DIN;
    float* Lpart = Mpart + (size_t)BB * CPB * DQ;

    const size_t ldsBytes = (size_t)LDS_FLOATS * sizeof(float);  // ~300 KB of the 320 KB WGP LDS

    attnpool_chunk_kernel<<<dim3(BB * CPB), dim3(256), ldsBytes, stream>>>(
        x, w, Opart, Mpart, Lpart);
    attnpool_combine_kernel<<<dim3(BB * DQ), dim3(128), 0, stream>>>(
        Opart, Mpart, Lpart, out);
}